// DummyGCN4_3745211482886
// MI455X (gfx1250) — compile-verified
//
#include <hip/hip_runtime.h>
#include <hip/hip_bf16.h>

// GCN4 on gfx1250 (MI455X):
//  - edge aggregation: fp32 global atomics (L2-resident, dominant cost),
//    wave-per-edge, float2 per lane.
//  - dense GEMMs (50000x64x128, 50000x128x64): V_WMMA_F32_16X16X4_F32,
//    one wave32 per 16x16 fp32 output tile, K/N compile-time so all inner-loop
//    loads use immediate offsets, K fully unrolled.
// n = 50000 = 3125*16, so all M tiles are full and EXEC is all-ones at WMMA.

typedef __attribute__((ext_vector_type(2))) float v2f;
typedef __attribute__((ext_vector_type(8))) float v8f;

#define NEG_SLOPE 0.01f

__device__ __forceinline__ float lrelu(float v) {
    return v >= 0.0f ? v : NEG_SLOPE * v;
}

// ---------------- elementwise / utility kernels ----------------

__global__ void zero_f32_kernel(float* __restrict__ p, int nelem) {
    int i = blockIdx.x * blockDim.x + threadIdx.x;
    if (i < nelem) p[i] = 0.0f;
}

// Layer-0 front half: s[dst] += in_feat[src]  (scalar features)
__global__ void agg_scalar_kernel(const float* __restrict__ x,
                                  const int* __restrict__ src,
                                  const int* __restrict__ dst,
                                  float* __restrict__ s, int ne) {
    int e = blockIdx.x * blockDim.x + threadIdx.x;
    if (e < ne) atomicAdd(&s[dst[e]], x[src[e]]);
}

// Layer-0 back half: h0[i][j] = lrelu(s[i]*W0[j] + b0[j])  (rank-1 expand, d=64)
__global__ void expand_l0_kernel(const float* __restrict__ s,
                                 const float* __restrict__ W0,
                                 const float* __restrict__ b0,
                                 float* __restrict__ h0, int n) {
    int gid = blockIdx.x * blockDim.x + threadIdx.x;
    int i = gid >> 6;
    int j = gid & 63;
    if (i < n) h0[gid] = lrelu(s[i] * W0[j] + b0[j]);
}

// 64-dim aggregation: wave per edge, lane handles dims {2*lane, 2*lane+1}.
// agg[dst[e]*64 + j] += x[src[e]*64 + j]
__global__ void agg_vec64_kernel(const float* __restrict__ x,
                                 const int* __restrict__ src,
                                 const int* __restrict__ dst,
                                 float* __restrict__ agg, int ne) {
    int gid = blockIdx.x * blockDim.x + threadIdx.x;
    int e = gid >> 5;
    int lane = gid & 31;
    if (e < ne) {
        int j0 = lane << 1;
        const float2 v = *(const float2*)(x + ((long long)src[e] << 6) + j0);
        float* d = agg + ((long long)dst[e] << 6) + j0;
        atomicAdd(d, v.x);
        atomicAdd(d + 1, v.y);
    }
}

// h[i][j] = lrelu(h[i][j] + b[j]), d = 64
__global__ void bias_act64_kernel(float* __restrict__ h, const float* __restrict__ b,
                                  int n) {
    int gid = blockIdx.x * blockDim.x + threadIdx.x;
    int i = gid >> 6;
    int j = gid & 63;
    if (i < n) h[gid] = lrelu(h[gid] + b[j]);
}

// t3[i] = dot(h2[i][0:64], W3[0:64])
__global__ void dot_w3_kernel(const float* __restrict__ h2,
                              const float* __restrict__ W3,
                              float* __restrict__ t3, int n) {
    int i = blockIdx.x * blockDim.x + threadIdx.x;
    if (i < n) {
        const float4* row = (const float4*)(h2 + ((long long)i << 6));
        const float4* w = (const float4*)W3;
        float acc = 0.0f;
#pragma unroll
        for (int k = 0; k < 16; ++k) {
            float4 r = row[k], ww = w[k];
            acc += r.x * ww.x + r.y * ww.y + r.z * ww.z + r.w * ww.w;
        }
        t3[i] = acc;
    }
}

// scalar += t3[src[e]] for edges with dst[e]==1 (only node 1's output is needed)
__global__ void final_agg_kernel(const float* __restrict__ t3,
                                 const int* __restrict__ src,
                                 const int* __restrict__ dst,
                                 float* __restrict__ scalar, int ne) {
    int e = blockIdx.x * blockDim.x + threadIdx.x;
    if (e < ne && dst[e] == 1) atomicAdd(scalar, t3[src[e]]);
}

__global__ void final_out_kernel(const float* __restrict__ scalar,
                                 const float* __restrict__ b3,
                                 float* __restrict__ out) {
    if (threadIdx.x == 0 && blockIdx.x == 0) out[0] = lrelu(scalar[0] + b3[0]);
}

// ---------------- WMMA fp32 GEMM ----------------
// out[n x N] = A[n x K] @ W[K x N]  (+bias, lrelu if APPLY)
// One wave32 per 16x16 output tile, K stepped by 4, fully unrolled.
// A-matrix layout (16x4 f32): lanes 0-15 K={k,k+1}, lanes 16-31 K={k+2,k+3}.
// C/D layout: lane<16 -> M=vgpr_i, N=lane ; lane>=16 -> M=8+vgpr_i, N=lane-16.
template <int K, int N, bool APPLY>
__global__ void gcn_gemm_wmma_kernel(const float* __restrict__ A,
                                     const float* __restrict__ W,
                                     const float* __restrict__ bias,
                                     float* __restrict__ out, int n) {
    int wave = (blockIdx.x * blockDim.x + threadIdx.x) >> 5;
    int lane = threadIdx.x & 31;
    constexpr int NTILES = N >> 4;
    int mtile = wave / NTILES;
    int ntile = wave - mtile * NTILES;
    if (mtile * 16 >= n) return;  // wave-uniform exit; EXEC all-ones below

    int mrow = mtile * 16 + (lane & 15);       // A row for this lane
    int mrow_c = mrow < n ? mrow : n - 1;      // clamp (no-op when n%16==0)
    int col = ntile * 16 + (lane & 15);        // B/C column for this lane
    int khalf = (lane >> 4) << 1;              // 0 for lanes 0-15, 2 for 16-31

    // single base pointers; all inner-loop loads at compile-time offsets
    const float* abase = A + (long long)mrow_c * K + khalf;
    const float* bbase = W + (long long)khalf * N + col;

    v8f c = {0.f, 0.f, 0.f, 0.f, 0.f, 0.f, 0.f, 0.f};

#pragma unroll
    for (int k = 0; k < K; k += 4) {
        v2f a, b;
        a.x = abase[k];            // -> global_load_b64 (a.x, a.y adjacent)
        a.y = abase[k + 1];
        b.x = bbase[k * N];        // immediate offsets k*N*4, (k+1)*N*4
        b.y = bbase[(k + 1) * N];
        c = __builtin_amdgcn_wmma_f32_16x16x4_f32(
                /*neg_a=*/false, a, /*neg_b=*/false, b,
                /*c_mod=*/(short)0, c, /*reuse_a=*/false, /*reuse_b=*/false);
    }

    int rbase = mtile * 16 + ((lane >> 4) << 3);   // M=0..7 or M=8..15 half
    float bv = APPLY ? bias[col] : 0.0f;
    float* obase = out + (long long)rbase * N + col;
#pragma unroll
    for (int i = 0; i < 8; ++i) {
        if (rbase + i < n) {
            float v = c[i] + bv;
            if (APPLY) v = lrelu(v);
            obase[i * N] = v;      // immediate offsets i*N*4
        }
    }
}

// ---------------- host launcher ----------------

static inline int cdiv(int a, int b) { return (a + b - 1) / b; }

extern "C" void kernel_launch(void* const* d_in, const int* in_sizes, int n_in,
                              void* d_out, int out_size, void* d_ws, size_t ws_size,
                              hipStream_t stream) {
    const float* in_feat = (const float*)d_in[0];
    const int*   src     = (const int*)d_in[1];
    const int*   dst     = (const int*)d_in[2];
    const float* W0 = (const float*)d_in[3];
    const float* b0 = (const float*)d_in[4];
    const float* W1 = (const float*)d_in[5];
    const float* b1 = (const float*)d_in[6];
    const float* W2 = (const float*)d_in[7];
    const float* b2 = (const float*)d_in[8];
    const float* W3 = (const float*)d_in[9];
    const float* b3 = (const float*)d_in[10];
    float* out = (float*)d_out;

    const int n  = in_sizes[0];   // 50000 (in_feat is [n,1])
    const int ne = in_sizes[1];   // 800000

    // workspace layout (floats): P0[n] | P1[64n] | P2[64n] | P3[128n] | P4[1]
    float* P0 = (float*)d_ws;                      // s0, later t3
    float* P1 = P0 + n;                            // h0, later t2
    float* P2 = P1 + (size_t)64 * n;               // agg1, later agg2/h2
    float* P3 = P2 + (size_t)64 * n;               // h1
    float* P4 = P3 + (size_t)128 * n;              // scalar accumulator

    const int B = 256;
    const int mtiles = cdiv(n, 16);                // 3125

    // ---- layer 0: scalar aggregate, rank-1 expand ----
    zero_f32_kernel<<<cdiv(n, B), B, 0, stream>>>(P0, n);
    agg_scalar_kernel<<<cdiv(ne, B), B, 0, stream>>>(in_feat, src, dst, P0, ne);
    expand_l0_kernel<<<cdiv(64 * n, B), B, 0, stream>>>(P0, W0, b0, P1, n);

    // ---- layer 1: aggregate 64-d, GEMM [n,64]@[64,128] + b1 + lrelu ----
    zero_f32_kernel<<<cdiv(64 * n, B), B, 0, stream>>>(P2, 64 * n);
    agg_vec64_kernel<<<cdiv(ne * 32, B), B, 0, stream>>>(P1, src, dst, P2, ne);
    {
        int waves = mtiles * (128 / 16);
        gcn_gemm_wmma_kernel<64, 128, true><<<cdiv(waves * 32, B), B, 0, stream>>>(
            P2, W1, b1, P3, n);
    }

    // ---- layer 2: GEMM [n,128]@[128,64] first, then aggregate, bias, lrelu ----
    {
        int waves = mtiles * (64 / 16);
        gcn_gemm_wmma_kernel<128, 64, false><<<cdiv(waves * 32, B), B, 0, stream>>>(
            P3, W2, nullptr, P1, n);
    }
    zero_f32_kernel<<<cdiv(64 * n, B), B, 0, stream>>>(P2, 64 * n);
    agg_vec64_kernel<<<cdiv(ne * 32, B), B, 0, stream>>>(P1, src, dst, P2, ne);
    bias_act64_kernel<<<cdiv(64 * n, B), B, 0, stream>>>(P2, b2, n);

    // ---- layer 3: dot with W3, aggregate only edges into node 1, lrelu ----
    dot_w3_kernel<<<cdiv(n, B), B, 0, stream>>>(P2, W3, P0, n);
    zero_f32_kernel<<<1, 32, 0, stream>>>(P4, 1);
    final_agg_kernel<<<cdiv(ne, B), B, 0, stream>>>(P0, src, dst, P4, ne);
    final_out_kernel<<<1, 32, 0, stream>>>(P4, b3, out);
}